// node_o_89404039233800
// MI455X (gfx1250) — compile-verified
//
#include <hip/hip_runtime.h>
#include <math.h>

typedef float v2f __attribute__((ext_vector_type(2)));
typedef float v8f __attribute__((ext_vector_type(8)));

#define T_DIM 1024
#define N_DIM 1024
#define INV_DT 2.0f
#define DIS_THRESH 12.0f

// One wave32 per tile of (fixed t, 16 consecutive agents n).
// A-fragment (16x4 f32): lane holds feat[m = lane%16][K = 2*(lane>=16) + {0,1}] in 2 VGPRs.
// B-fragment (4x16 f32): slice of W6^T, K = vgpr + 2*lane_half, N = lane%16 (+16j block).
// C = bias b6 broadcast; D = 16x16 out tile via v_wmma_f32_16x16x4_f32.
__global__ __launch_bounds__(256) void fused_traj_wmma_kernel(
    const float* __restrict__ a,
    const float* __restrict__ W1, const float* __restrict__ b1,
    const float* __restrict__ W2, const float* __restrict__ b2,
    const float* __restrict__ W3, const float* __restrict__ b3,
    const float* __restrict__ W4, const float* __restrict__ b4,
    const float* __restrict__ W5, const float* __restrict__ b5,
    const float* __restrict__ W6, const float* __restrict__ b6,
    float* __restrict__ out1, float* __restrict__ out64)
{
    const int lane = threadIdx.x & 31;
    // Broadcast the tile id so t / nb are provably wave-uniform (SGPRs):
    // -> scalar branch for the t==0 duplicate row, SALU addressing, SMEM joint loads.
    const int tile = __builtin_amdgcn_readfirstlane(
        (int)(blockIdx.x * 8 + (threadIdx.x >> 5)));   // 0 .. 65471, exact fit
    const int t    = tile >> 6;               // 0..1022 (source timestep; output row t+1)
    const int nb   = (tile & 63) << 4;        // agent tile base
    const int half = lane >> 4;               // 0: K={0,1}, 1: K={2,3}
    const int m    = lane & 15;               // row within tile
    const int n    = nb + m;

    // ---- pointwise feature computation (each lane: one agent row) ----
    const float* ap    = a + ((size_t)(t + 1) * N_DIM + n) * 2;
    const float* aprev = a + ((size_t)t       * N_DIM + n) * 2;
    const float px = ap[0],    py = ap[1];
    const float qx = aprev[0], qy = aprev[1];

    // joint (last) agent: uniform address -> scalar loads
    const float* jp    = a + ((size_t)(t + 1) * N_DIM + (N_DIM - 1)) * 2;
    const float* jprev = a + ((size_t)t       * N_DIM + (N_DIM - 1)) * 2;
    const float pjx = jp[0],    pjy = jp[1];
    const float qjx = jprev[0], qjy = jprev[1];

    const float dsx = px - pjx, dsy = py - pjy;
    const float dis = sqrtf(dsx * dsx + dsy * dsy);
    const float ddx = px - qx,  ddy = py - qy;
    const float v   = sqrtf(ddx * ddx + ddy * ddy) * INV_DT;
    const float ang = atan2f(ddy, ddx);
    const float djx = pjx - qjx, djy = pjy - qjy;
    const float vj   = sqrtf(djx * djx + djy * djy) * INV_DT;
    const float angj = atan2f(djy, djx);
    const float mask = (dis <= DIS_THRESH) ? 1.0f : 0.0f;

    const float x_m = W1[0] * px  + W1[1] * (mask * pjx)  + b1[0];
    const float y_m = W2[0] * py  + W2[1] * (mask * pjy)  + b2[0];
    const float v_m = W3[0] * v   + W3[1] * (mask * vj)   + b3[0];
    const float a_m = W4[0] * ang + W4[1] * (mask * angj) + b4[0];

    // 4->1 head (VALU dot; trivially cheap)
    const float s5 = W5[0] * x_m + W5[1] * y_m + W5[2] * v_m + W5[3] * a_m + b5[0];

    // ---- A fragment: K = vgpr + 2*half ----
    v2f A;
    A.x = half ? v_m : x_m;   // K = 2*half
    A.y = half ? a_m : y_m;   // K = 2*half + 1

    // Uniform row bases (SGPR); per-lane offset is the small in-tile part.
    float* const rowp  = out64 + ((size_t)(t + 1) * N_DIM + nb) * 64;
    float* const rowp0 = out64 + (size_t)nb * 64;
    const int laneoff  = 8 * half * 64 + m;   // row = 8*half (+r), col = m (+16j)

    // ---- 4 WMMA tiles cover the 64 output channels ----
    #pragma unroll
    for (int j = 0; j < 4; ++j) {
        const int o = j * 16 + m;         // output channel for this lane column
        v2f B;
        B.x = W6[o * 4 + 2 * half + 0];   // K = 2*half
        B.y = W6[o * 4 + 2 * half + 1];   // K = 2*half + 1
        const float bias = b6[o];
        v8f C;
        #pragma unroll
        for (int r = 0; r < 8; ++r) C[r] = bias;

        // 8 args: (neg_a, A, neg_b, B, c_mod, C, reuse_a, reuse_b)
        v8f D = __builtin_amdgcn_wmma_f32_16x16x4_f32(
            false, A, false, B, (short)0, C, false, false);

        // D[r] = out64[row = r + 8*half][col = o]; 16 lanes -> 16 contiguous channels
        #pragma unroll
        for (int r = 0; r < 8; ++r) {
            rowp[laneoff + j * 16 + r * 64] = D[r];
        }
        if (t == 0) {   // scalar branch now (t is SGPR): duplicate into output row 0
            #pragma unroll
            for (int r = 0; r < 8; ++r) {
                rowp0[laneoff + j * 16 + r * 64] = D[r];
            }
        }
    }

    // ---- out1 (after all WMMAs; EXEC divergence is safe here) ----
    if (half == 0) {
        out1[(size_t)(t + 1) * N_DIM + n] = s5;
        if (t == 0) out1[n] = s5;
    }
}

extern "C" void kernel_launch(void* const* d_in, const int* in_sizes, int n_in,
                              void* d_out, int out_size, void* d_ws, size_t ws_size,
                              hipStream_t stream) {
    const float* a  = (const float*)d_in[0];
    const float* W1 = (const float*)d_in[1];
    const float* b1 = (const float*)d_in[2];
    const float* W2 = (const float*)d_in[3];
    const float* b2 = (const float*)d_in[4];
    const float* W3 = (const float*)d_in[5];
    const float* b3 = (const float*)d_in[6];
    const float* W4 = (const float*)d_in[7];
    const float* b4 = (const float*)d_in[8];
    const float* W5 = (const float*)d_in[9];
    const float* b5 = (const float*)d_in[10];
    const float* W6 = (const float*)d_in[11];
    const float* b6 = (const float*)d_in[12];

    float* out1  = (float*)d_out;                                 // (1024,1024,1)
    float* out64 = (float*)d_out + (size_t)T_DIM * N_DIM;         // (1024,1024,64)

    // 1023 timesteps * 64 agent-tiles = 65472 waves; 8 waves/block -> 8184 blocks
    fused_traj_wmma_kernel<<<8184, 256, 0, stream>>>(
        a, W1, b1, W2, b2, W3, b3, W4, b4, W5, b5, W6, b6, out1, out64);
}